// VINNetwork_31602369364686
// MI455X (gfx1250) — compile-verified
//
#include <hip/hip_runtime.h>

typedef __attribute__((ext_vector_type(2))) float v2f;
typedef __attribute__((ext_vector_type(8))) float v8f;

#define GH 4
#define GW 4
#define GC 3
#define NA 4
#define NK 20
#define BLOCK 128  // 4 wave32 waves; 128 samples per block

__global__ __launch_bounds__(BLOCK) void vin_fused_kernel(
    const float* __restrict__ obs,    // (B, 48)
    const float* __restrict__ PhiW,   // (48, 48)
    const float* __restrict__ PhiB,   // (48)
    const float* __restrict__ LW,     // (4, 36)
    const float* __restrict__ LB,     // (4)
    float* __restrict__ out,          // (B, 4)
    int B)
{
    __shared__ float sObs[BLOCK * 48];   // staged observations
    __shared__ float sOut[BLOCK * 48];   // Phi output (rin/rout/p interleaved)
    __shared__ float sPhiW[48 * 48];
    __shared__ float sPhiB[48];
    __shared__ float sLW[NA * 36];
    __shared__ float sLB[NA];

    const int tid = threadIdx.x;
    const int blockBase = blockIdx.x * BLOCK;  // first sample of this block

    // ---------- stage obs tile (coalesced float4, zero-fill past end) ----------
    {
        const float4* g4 = reinterpret_cast<const float4*>(obs);
        float4* s4 = reinterpret_cast<float4*>(sObs);
        const int total4 = B * 12;           // B*48/4
        const int base4 = blockBase * 12;
        #pragma unroll
        for (int i = 0; i < 12; ++i) {
            int g = base4 + tid + i * BLOCK;
            float4 val = make_float4(0.f, 0.f, 0.f, 0.f);
            if (g < total4) val = g4[g];
            s4[tid + i * BLOCK] = val;
        }
    }
    // ---------- stage weights ----------
    for (int i = tid; i < 48 * 48; i += BLOCK) sPhiW[i] = PhiW[i];
    if (tid < 48) sPhiB[tid] = PhiB[tid];
    for (int i = tid; i < NA * 36; i += BLOCK) sLW[i] = LW[i];
    if (tid < NA) sLB[tid] = LB[tid];
    __syncthreads();

    // ---------- Phase 1: out = obs @ PhiW^T + PhiB via V_WMMA_F32_16X16X4_F32 ----------
    // Each wave handles 32 samples as two 16-row M-tiles; N=48 -> 3 tiles; K=48 -> 12 steps.
    const int lane = tid & 31;
    const int wave = tid >> 5;
    const int m  = lane & 15;   // A: row M / B,D: column N (mod 16)
    const int hi = lane >> 4;   // K-half selector for A/B fragments; +8 row for D

    for (int mt = 0; mt < 2; ++mt) {
        const int srow = wave * 32 + mt * 16 + m;   // sample row in block tile
        v8f acc0 = {}, acc1 = {}, acc2 = {};
        for (int ks = 0; ks < 12; ++ks) {
            const int kk = ks * 4 + hi * 2;
            v2f a;
            a.x = sObs[srow * 48 + kk];
            a.y = sObs[srow * 48 + kk + 1];
            v2f b0, b1, b2;  // B[k][n] = PhiW[n][k]
            b0.x = sPhiW[(0  + m) * 48 + kk];  b0.y = sPhiW[(0  + m) * 48 + kk + 1];
            b1.x = sPhiW[(16 + m) * 48 + kk];  b1.y = sPhiW[(16 + m) * 48 + kk + 1];
            b2.x = sPhiW[(32 + m) * 48 + kk];  b2.y = sPhiW[(32 + m) * 48 + kk + 1];
            acc0 = __builtin_amdgcn_wmma_f32_16x16x4_f32(false, a, false, b0, (short)0, acc0, false, false);
            acc1 = __builtin_amdgcn_wmma_f32_16x16x4_f32(false, a, false, b1, (short)0, acc1, false, false);
            acc2 = __builtin_amdgcn_wmma_f32_16x16x4_f32(false, a, false, b2, (short)0, acc2, false, false);
        }
        // D layout: VGPR r holds M = r + 8*hi, N = lane%16. Scatter + bias into sOut.
        #pragma unroll
        for (int r = 0; r < 8; ++r) {
            const int row = wave * 32 + mt * 16 + r + 8 * hi;
            sOut[row * 48 + 0  + m] = acc0[r] + sPhiB[0  + m];
            sOut[row * 48 + 16 + m] = acc1[r] + sPhiB[16 + m];
            sOut[row * 48 + 32 + m] = acc2[r] + sPhiB[32 + m];
        }
    }
    __syncthreads();

    // ---------- Phase 2: per-sample K=20 value iteration (one sample per lane) ----------
    float rin[16], rou[16], pp[16], v[16];
    #pragma unroll
    for (int h = 0; h < GH; ++h)
        #pragma unroll
        for (int w = 0; w < GW; ++w) {
            const int f = h * (GW * GC) + w * GC;
            rin[h * 4 + w] = sOut[tid * 48 + f + 0];
            rou[h * 4 + w] = sOut[tid * 48 + f + 1];
            pp [h * 4 + w] = sOut[tid * 48 + f + 2];
            v  [h * 4 + w] = 0.f;
        }

    #pragma unroll 1
    for (int k = 0; k < NK; ++k) {
        float vp[16];
        #pragma unroll
        for (int i = 0; i < 16; ++i) vp[i] = v[i];  // snapshot at step start
        // OFFSETS [(0,1),(2,1),(1,0),(1,2)] -> neighbor deltas (dh,dw):
        // (0,-1), (0,+1), (-1,0), (+1,0)
        #pragma unroll
        for (int o = 0; o < 4; ++o) {
            const int dh = (o == 2) ? -1 : ((o == 3) ? 1 : 0);
            const int dw = (o == 0) ? -1 : ((o == 1) ? 1 : 0);
            #pragma unroll
            for (int h = 0; h < GH; ++h)
                #pragma unroll
                for (int w = 0; w < GW; ++w) {
                    const int nh = h + dh, nw = w + dw;
                    const bool in = ((unsigned)nh < (unsigned)GH) && ((unsigned)nw < (unsigned)GW);
                    const float vn = in ? vp [nh * 4 + nw] : 0.f;
                    const float rn = in ? rin[nh * 4 + nw] : 0.f;
                    const float nv = fmaf(pp[h * 4 + w], vn, rn - rou[h * 4 + w]);
                    v[h * 4 + w] = fmaxf(v[h * 4 + w], nv);
                }
        }
    }

    // ---------- Phase 3: agent position, 3x3 gather, logit GEMV ----------
    int pos = -1;
    #pragma unroll
    for (int idx = 0; idx < 16; ++idx) {
        if (pos < 0 && sObs[tid * 48 + idx * 3 + 1] != 0.f) pos = idx;
    }
    if (pos < 0) pos = 5;  // default (1,1)
    const int pi = pos >> 2, pj = pos & 3;

    float lg0 = sLB[0], lg1 = sLB[1], lg2 = sLB[2], lg3 = sLB[3];
    #pragma unroll
    for (int di = 0; di < 3; ++di)
        #pragma unroll
        for (int dj = 0; dj < 3; ++dj) {
            const int r = pi + di - 1, c = pj + dj - 1;
            const bool in = ((unsigned)r < (unsigned)GH) && ((unsigned)c < (unsigned)GW);
            const int cell = in ? (r * 4 + c) : 0;
            #pragma unroll
            for (int ch = 0; ch < GC; ++ch) {
                const float val = in ? sObs[tid * 48 + cell * 3 + ch] : 0.f;
                const int t = (di * 3 + dj) * 3 + ch;
                lg0 = fmaf(val, sLW[0 * 36 + t], lg0);
                lg1 = fmaf(val, sLW[1 * 36 + t], lg1);
                lg2 = fmaf(val, sLW[2 * 36 + t], lg2);
                lg3 = fmaf(val, sLW[3 * 36 + t], lg3);
            }
            const float sv = in ? v[cell] : 0.f;
            const int t = 27 + di * 3 + dj;
            lg0 = fmaf(sv, sLW[0 * 36 + t], lg0);
            lg1 = fmaf(sv, sLW[1 * 36 + t], lg1);
            lg2 = fmaf(sv, sLW[2 * 36 + t], lg2);
            lg3 = fmaf(sv, sLW[3 * 36 + t], lg3);
        }

    const int s = blockBase + tid;
    if (s < B) {
        out[s * 4 + 0] = lg0;
        out[s * 4 + 1] = lg1;
        out[s * 4 + 2] = lg2;
        out[s * 4 + 3] = lg3;
    }
}

extern "C" void kernel_launch(void* const* d_in, const int* in_sizes, int n_in,
                              void* d_out, int out_size, void* d_ws, size_t ws_size,
                              hipStream_t stream) {
    (void)n_in; (void)out_size; (void)d_ws; (void)ws_size;
    const float* obs  = (const float*)d_in[0];
    const float* PhiW = (const float*)d_in[1];
    const float* PhiB = (const float*)d_in[2];
    const float* LW   = (const float*)d_in[3];
    const float* LB   = (const float*)d_in[4];
    float* out = (float*)d_out;

    const int B = in_sizes[0] / 48;
    const int grid = (B + BLOCK - 1) / BLOCK;
    vin_fused_kernel<<<grid, BLOCK, 0, stream>>>(obs, PhiW, PhiB, LW, LB, out, B);
}